// LowpassDetector_42691974922776
// MI455X (gfx1250) — compile-verified
//
#include <hip/hip_runtime.h>
#include <hip/hip_bf16.h>

typedef __attribute__((ext_vector_type(2))) float v2f;
typedef __attribute__((ext_vector_type(8))) float v8f;

// Problem geometry (reference: signal [16384, 2048] float32, order-4 IIR)
constexpr int S       = 2048;   // channels
constexpr int T       = 16384;  // time samples
constexpr int L       = 16;     // chunk length (== WMMA M/N)
constexpr int NCHUNK  = T / L;  // 1024
constexpr int NGROUP  = S / 16; // 128 channel groups
constexpr int SBLK    = 32;     // chunks per superblock
constexpr int NSB     = NCHUNK / SBLK; // 32 superblocks

// Workspace layout (floats)
constexpr size_t OFF_H    = 0;                       // H[16][16]
constexpr size_t OFF_P    = 256;                     // P[16][4]
constexpr size_t OFF_Q    = 320;                     // Qpad[16][16]
constexpr size_t OFF_APOW = 576;                     // apow[i]=A16^i, i=0..32 (33*16)
constexpr size_t OFF_ZC   = 4096;                    // Zc / (after scan1) w : [NCHUNK][4][S]
constexpr size_t OFF_U    = OFF_ZC + (size_t)NCHUNK * 4 * S;  // u: [NSB][4][S]
constexpr size_t OFF_ZS   = OFF_U  + (size_t)NSB    * 4 * S;  // zs: [NSB][4][S]

// ---------------------------------------------------------------------------
// Setup: derive state-space constants from (b, a) on device. Single thread.
// State update (transposed DF-II, matches reference step()):
//   y = b0*x + z[0]
//   z'[i] = z[i+1] - a[i+1]*z[0] + (b[i+1] - a[i+1]*b0)*x
// => A[i][j] = [j==i+1] - a[i+1]*[j==0],  g[i] = b[i+1]-a[i+1]*b0, c=e0, d=b0
// ---------------------------------------------------------------------------
__global__ void lp_setup_kernel(const float* __restrict__ bc,
                                const float* __restrict__ ac,
                                float* __restrict__ ws) {
  if (threadIdx.x != 0 || blockIdx.x != 0) return;
  const float b0 = bc[0];
  float A[4][4], g4[4];
  for (int i = 0; i < 4; ++i) {
    for (int j = 0; j < 4; ++j) A[i][j] = (j == i + 1) ? 1.0f : 0.0f;
    A[i][0] -= ac[i + 1];
    g4[i] = bc[i + 1] - ac[i + 1] * b0;
  }
  // Impulse response h[0..15]: h[0]=b0, h[m]=(A^{m-1} g)[0]
  float h[16];
  h[0] = b0;
  float v[4] = {g4[0], g4[1], g4[2], g4[3]};
  for (int t = 1; t < 16; ++t) {
    h[t] = v[0];
    float nv[4];
    for (int i = 0; i < 4; ++i) {
      float s = 0.f;
      for (int j = 0; j < 4; ++j) s += A[i][j] * v[j];
      nv[i] = s;
    }
    for (int i = 0; i < 4; ++i) v[i] = nv[i];
  }
  // H: lower-triangular Toeplitz
  for (int t = 0; t < 16; ++t)
    for (int k = 0; k < 16; ++k)
      ws[OFF_H + t * 16 + k] = (k <= t) ? h[t - k] : 0.0f;
  // P[t][:] = row 0 of A^t ; also build A16 = A^16
  float M[4][4];
  for (int i = 0; i < 4; ++i)
    for (int j = 0; j < 4; ++j) M[i][j] = (i == j) ? 1.0f : 0.0f;
  for (int t = 0; t < 16; ++t) {
    for (int m = 0; m < 4; ++m) ws[OFF_P + t * 4 + m] = M[0][m];
    float NM[4][4];
    for (int i = 0; i < 4; ++i)
      for (int j = 0; j < 4; ++j) {
        float s = 0.f;
        for (int k = 0; k < 4; ++k) s += A[i][k] * M[k][j];
        NM[i][j] = s;
      }
    for (int i = 0; i < 4; ++i)
      for (int j = 0; j < 4; ++j) M[i][j] = NM[i][j];
  }
  float A16[4][4];
  for (int i = 0; i < 4; ++i)
    for (int j = 0; j < 4; ++j) A16[i][j] = M[i][j];
  // Qpad: rows 0..3 = Q, Q[:,k] = A^{15-k} g ; rows 4..15 zero
  for (int i = 0; i < 256; ++i) ws[OFF_Q + i] = 0.0f;
  for (int i = 0; i < 4; ++i) v[i] = g4[i];
  for (int k = 15; k >= 0; --k) {
    for (int m = 0; m < 4; ++m) ws[OFF_Q + m * 16 + k] = v[m];
    float nv[4];
    for (int i = 0; i < 4; ++i) {
      float s = 0.f;
      for (int j = 0; j < 4; ++j) s += A[i][j] * v[j];
      nv[i] = s;
    }
    for (int i = 0; i < 4; ++i) v[i] = nv[i];
  }
  // apow[i] = A16^i, i = 0..32
  float Pw[4][4];
  for (int i = 0; i < 4; ++i)
    for (int j = 0; j < 4; ++j) Pw[i][j] = (i == j) ? 1.0f : 0.0f;
  for (int p = 0; p <= 32; ++p) {
    for (int i = 0; i < 4; ++i)
      for (int j = 0; j < 4; ++j) ws[OFF_APOW + p * 16 + i * 4 + j] = Pw[i][j];
    float NM[4][4];
    for (int i = 0; i < 4; ++i)
      for (int j = 0; j < 4; ++j) {
        float s = 0.f;
        for (int k = 0; k < 4; ++k) s += Pw[i][k] * A16[k][j];
        NM[i][j] = s;
      }
    for (int i = 0; i < 4; ++i)
      for (int j = 0; j < 4; ++j) Pw[i][j] = NM[i][j];
  }
}

// ---------------------------------------------------------------------------
// Pass 1: Zc = Q @ X per (chunk, channel-group) tile via f32 WMMA.
// One wave per 16x16 tile; blockDim = 256 (8 waves).
// ---------------------------------------------------------------------------
__global__ __launch_bounds__(256) void lp_zc_kernel(const float* __restrict__ x,
                                                    float* __restrict__ ws) {
  const int wave = threadIdx.x >> 5;
  const int lane = threadIdx.x & 31;
  const int tile = blockIdx.x * 8 + wave;
  const int c = tile >> 7;       // chunk 0..1023
  const int g = tile & 127;      // channel group
  const int half = lane >> 4;    // lane half selects K pair
  const int l16 = lane & 15;
  const float* Q = ws + OFF_Q;
  const float* xt = x + (size_t)c * L * S + (size_t)g * 16 + l16;
  v8f acc = {};
#pragma unroll
  for (int k = 0; k < 4; ++k) {
    const int kk0 = 4 * k + 2 * half;
    v2f av, bv;
    av[0] = Q[l16 * 16 + kk0];
    av[1] = Q[l16 * 16 + kk0 + 1];
    bv[0] = xt[(size_t)kk0 * S];
    bv[1] = xt[(size_t)(kk0 + 1) * S];
    acc = __builtin_amdgcn_wmma_f32_16x16x4_f32(false, av, false, bv,
                                                (short)0, acc, false, false);
  }
  if (half == 0) {  // rows 0..3 of D are the 4 state components
    float* zc = ws + OFF_ZC + (size_t)c * 4 * S + (size_t)g * 16 + l16;
    zc[0 * (size_t)S] = acc[0];
    zc[1 * (size_t)S] = acc[1];
    zc[2 * (size_t)S] = acc[2];
    zc[3 * (size_t)S] = acc[3];
  }
}

// ---------------------------------------------------------------------------
// Pass 2a: within each superblock, sequential scan over 32 chunks.
// In-place: Zc[c] is replaced by w[c] = incoming state of chunk c (local).
// One thread per (channel, superblock): 65536 threads.
// ---------------------------------------------------------------------------
__global__ __launch_bounds__(256) void lp_scan1_kernel(float* __restrict__ ws) {
  const int tid = blockIdx.x * 256 + threadIdx.x;
  const int s  = tid & (S - 1);
  const int sb = tid >> 11;
  const float* A16 = ws + OFF_APOW + 16;  // apow[1]
  float a[16];
#pragma unroll
  for (int i = 0; i < 16; ++i) a[i] = A16[i];
  float w0 = 0.f, w1 = 0.f, w2 = 0.f, w3 = 0.f;
  float* zc = ws + OFF_ZC + ((size_t)sb * SBLK) * 4 * S + s;
  for (int cc = 0; cc < SBLK; ++cc) {
    const float r0 = zc[0 * (size_t)S], r1 = zc[1 * (size_t)S];
    const float r2 = zc[2 * (size_t)S], r3 = zc[3 * (size_t)S];
    zc[0 * (size_t)S] = w0; zc[1 * (size_t)S] = w1;
    zc[2 * (size_t)S] = w2; zc[3 * (size_t)S] = w3;
    const float n0 = a[0]*w0 + a[1]*w1 + a[2]*w2 + a[3]*w3 + r0;
    const float n1 = a[4]*w0 + a[5]*w1 + a[6]*w2 + a[7]*w3 + r1;
    const float n2 = a[8]*w0 + a[9]*w1 + a[10]*w2 + a[11]*w3 + r2;
    const float n3 = a[12]*w0 + a[13]*w1 + a[14]*w2 + a[15]*w3 + r3;
    w0 = n0; w1 = n1; w2 = n2; w3 = n3;
    zc += (size_t)4 * S;
  }
  float* u = ws + OFF_U + (size_t)sb * 4 * S + s;
  u[0 * (size_t)S] = w0; u[1 * (size_t)S] = w1;
  u[2 * (size_t)S] = w2; u[3 * (size_t)S] = w3;
}

// ---------------------------------------------------------------------------
// Pass 2b: scan across 32 superblocks per channel: zs_{k+1} = A16^32 zs_k + u_k
// ---------------------------------------------------------------------------
__global__ __launch_bounds__(256) void lp_scan2_kernel(float* __restrict__ ws) {
  const int s = blockIdx.x * 256 + threadIdx.x;  // 0..2047
  const float* A = ws + OFF_APOW + 32 * 16;      // apow[32] = A16^32
  float a[16];
#pragma unroll
  for (int i = 0; i < 16; ++i) a[i] = A[i];
  float z0 = 0.f, z1 = 0.f, z2 = 0.f, z3 = 0.f;
  for (int sb = 0; sb < NSB; ++sb) {
    float* zs = ws + OFF_ZS + (size_t)sb * 4 * S + s;
    zs[0 * (size_t)S] = z0; zs[1 * (size_t)S] = z1;
    zs[2 * (size_t)S] = z2; zs[3 * (size_t)S] = z3;
    const float* u = ws + OFF_U + (size_t)sb * 4 * S + s;
    const float u0 = u[0 * (size_t)S], u1 = u[1 * (size_t)S];
    const float u2 = u[2 * (size_t)S], u3 = u[3 * (size_t)S];
    const float n0 = a[0]*z0 + a[1]*z1 + a[2]*z2 + a[3]*z3 + u0;
    const float n1 = a[4]*z0 + a[5]*z1 + a[6]*z2 + a[7]*z3 + u1;
    const float n2 = a[8]*z0 + a[9]*z1 + a[10]*z2 + a[11]*z3 + u2;
    const float n3 = a[12]*z0 + a[13]*z1 + a[14]*z2 + a[15]*z3 + u3;
    z0 = n0; z1 = n1; z2 = n2; z3 = n3;
  }
}

// ---------------------------------------------------------------------------
// Pass 3: Y = H @ X + P @ z_in per tile. z_in = A16^(c%32) @ zs[sb] + w[c].
// 5 f32 WMMAs per tile (1 for P correction, 4 for the Toeplitz GEMM).
// ---------------------------------------------------------------------------
__global__ __launch_bounds__(256) void lp_final_kernel(const float* __restrict__ x,
                                                       float* __restrict__ out,
                                                       const float* __restrict__ ws) {
  const int wave = threadIdx.x >> 5;
  const int lane = threadIdx.x & 31;
  const int tile = blockIdx.x * 8 + wave;
  const int c = tile >> 7;
  const int g = tile & 127;
  const int half = lane >> 4;
  const int l16 = lane & 15;
  const int sb = c >> 5;
  const int cc = c & 31;
  const int chan = g * 16 + l16;

  // z_in for this lane's two B-rows (K = 2*half + j)
  const float* zs = ws + OFF_ZS + (size_t)sb * 4 * S + chan;
  const float z0 = zs[0 * (size_t)S], z1 = zs[1 * (size_t)S];
  const float z2 = zs[2 * (size_t)S], z3 = zs[3 * (size_t)S];
  const float* ap = ws + OFF_APOW + cc * 16;
  const float* W  = ws + OFF_ZC + (size_t)c * 4 * S + chan;
  v2f zin;
#pragma unroll
  for (int j = 0; j < 2; ++j) {
    const int K = 2 * half + j;
    zin[j] = ap[K * 4 + 0] * z0 + ap[K * 4 + 1] * z1 +
             ap[K * 4 + 2] * z2 + ap[K * 4 + 3] * z3 + W[(size_t)K * S];
  }
  const float* Pm = ws + OFF_P;
  v2f pa;
  pa[0] = Pm[l16 * 4 + 2 * half + 0];
  pa[1] = Pm[l16 * 4 + 2 * half + 1];

  v8f acc = {};
  acc = __builtin_amdgcn_wmma_f32_16x16x4_f32(false, pa, false, zin,
                                              (short)0, acc, false, false);

  const float* H = ws + OFF_H;
  const float* xt = x + (size_t)c * L * S + chan;
#pragma unroll
  for (int k = 0; k < 4; ++k) {
    const int kk0 = 4 * k + 2 * half;
    v2f av, bv;
    av[0] = H[l16 * 16 + kk0];
    av[1] = H[l16 * 16 + kk0 + 1];
    bv[0] = xt[(size_t)kk0 * S];
    bv[1] = xt[(size_t)(kk0 + 1) * S];
    acc = __builtin_amdgcn_wmma_f32_16x16x4_f32(false, av, false, bv,
                                                (short)0, acc, false, false);
  }

  float* o = out + (size_t)c * L * S + chan;
#pragma unroll
  for (int r = 0; r < 8; ++r)
    o[(size_t)(r + 8 * half) * S] = acc[r];
}

// ---------------------------------------------------------------------------
extern "C" void kernel_launch(void* const* d_in, const int* in_sizes, int n_in,
                              void* d_out, int out_size, void* d_ws, size_t ws_size,
                              hipStream_t stream) {
  const float* signal = (const float*)d_in[0];  // [16384, 2048] f32
  const float* b      = (const float*)d_in[1];  // [5] f32
  const float* a      = (const float*)d_in[2];  // [5] f32
  float* out = (float*)d_out;
  float* ws  = (float*)d_ws;

  lp_setup_kernel<<<1, 64, 0, stream>>>(b, a, ws);

  const int tiles  = NCHUNK * NGROUP;       // 131072
  const int blocks = tiles / 8;             // 8 waves (tiles) per 256-thread block
  lp_zc_kernel<<<blocks, 256, 0, stream>>>(signal, ws);

  lp_scan1_kernel<<<(S * NSB) / 256, 256, 0, stream>>>(ws);
  lp_scan2_kernel<<<S / 256, 256, 0, stream>>>(ws);

  lp_final_kernel<<<blocks, 256, 0, stream>>>(signal, out, ws);
}